// DigitCapsules_35570919145496
// MI455X (gfx1250) — compile-verified
//
#include <hip/hip_runtime.h>
#include <hip/hip_bf16.h>
#include <math.h>

// Problem constants (from reference setup_inputs)
#define BATCH 128
#define RNUM  2048
#define ONUM  16
#define DNUM  16
#define CNUM  16

typedef __attribute__((ext_vector_type(2))) float v2f;
typedef __attribute__((ext_vector_type(8))) float v8f;

// Sum across each 16-lane half of the wave32 (xor masks 1,2,4,8 never cross
// the 16-lane boundary). All lanes in a half end with the half's sum.
__device__ __forceinline__ float halfsum16(float v) {
    v += __shfl_xor(v, 1);
    v += __shfl_xor(v, 2);
    v += __shfl_xor(v, 4);
    v += __shfl_xor(v, 8);
    return v;
}

// One wave handles one r and a tile of 16 batches.
// Block = 256 threads = 8 waves = all 8 batch-tiles of one r (W[r] shared via cache).
// Grid = 2048 blocks (one per r).
//
// MODE 0: c_ij uniform (1/R); writes b1 = agreement to bmat.
// MODE 1: c_ij = exp(b - rowmax)/denom; writes b2 = b1 + agreement (in place).
// MODE 2: same softmax; writes v_j to out (B,R,C).
template <int MODE>
__global__ __launch_bounds__(256) void caps_kernel(
    const float* __restrict__ x,        // (B, D)
    const float* __restrict__ W,        // (R, O, D, C)
    float* __restrict__ bmat,           // (B, R, O) routing logits
    const float* __restrict__ rowmax,   // (B, O)
    const float* __restrict__ denom,    // (B, O)
    float* __restrict__ out)            // (B, R, C)
{
    const int r    = blockIdx.x;
    const int tid  = threadIdx.x;
    const int L    = tid & 31;          // lane
    const int w    = tid >> 5;          // wave in block = batch tile
    const int h    = L >> 4;            // half-wave
    const int m16  = L & 15;
    const int bb0  = w * 16;            // first batch of this wave's tile

    // ---- A-matrix (X tile, 16x4 f32 per chunk), per ISA layout:
    // VGPR0: lanes0-15 K=4k+0, lanes16-31 K=4k+2 ; VGPR1: K=4k+1 / K=4k+3
    v2f a[4];
    const float* xb = x + (bb0 + m16) * DNUM;
    #pragma unroll
    for (int k = 0; k < 4; ++k) {
        a[k] = *reinterpret_cast<const v2f*>(xb + 4 * k + 2 * h);
    }

    // ---- u_hat tiles: U[o][i], lane L holds U[m = i + 8*h][c = m16]
    v8f U[ONUM];
    const float* Wr = W + (size_t)r * ONUM * DNUM * CNUM;
    #pragma unroll
    for (int o = 0; o < ONUM; ++o) {
        const float* Wo = Wr + o * (DNUM * CNUM);
        v8f acc = {0.f, 0.f, 0.f, 0.f, 0.f, 0.f, 0.f, 0.f};
        #pragma unroll
        for (int k = 0; k < 4; ++k) {
            // B-matrix 4x16 f32 chunk: VGPR j holds rows (4k+j) / (4k+j+2)
            v2f bmx;
            const int row0 = 4 * k + 2 * h;
            bmx.x = Wo[(row0 + 0) * CNUM + m16];
            bmx.y = Wo[(row0 + 1) * CNUM + m16];
            acc = __builtin_amdgcn_wmma_f32_16x16x4_f32(
                false, a[k], false, bmx, (short)0, acc, false, false);
        }
        U[o] = acc;
    }

    // ---- routing coefficients for this wave's 16 b x 16 o block
    // lane L, slot i  <->  (m = i + 8*h, o = m16)
    float braw[8], creg[8];
    #pragma unroll
    for (int i = 0; i < 8; ++i) {
        if (MODE == 0) {
            braw[i] = 0.0f;
            creg[i] = 1.0f / (float)RNUM;   // softmax of all-zeros over R
        } else {
            const int b = bb0 + i + 8 * h;
            braw[i] = bmat[((size_t)b * RNUM + r) * ONUM + m16];
            const float rm = rowmax[b * ONUM + m16];
            const float dn = denom[b * ONUM + m16];
            creg[i] = __expf(braw[i] - rm) / dn;
        }
    }

    // ---- s[b,c] = sum_o c[b,o] * u_hat[b,o,c]
    v8f s = {0.f, 0.f, 0.f, 0.f, 0.f, 0.f, 0.f, 0.f};
    #pragma unroll
    for (int o = 0; o < ONUM; ++o) {
        #pragma unroll
        for (int i = 0; i < 8; ++i) {
            float co;
            if (MODE == 0) {
                co = 1.0f / (float)RNUM;
            } else {
                // c[m = i + 8*h, o] lives in lane (h*16 + o), slot i
                co = __shfl(creg[i], (tid & 16) | o);
            }
            s[i] += co * U[o][i];
        }
    }

    // ---- squash over C: v = s * sqrt(ns) / (1 + ns)   (no epsilon, as in ref)
    v8f vj;
    #pragma unroll
    for (int i = 0; i < 8; ++i) {
        const float ns  = halfsum16(s[i] * s[i]);
        const float fac = sqrtf(ns) / (1.0f + ns);
        vj[i] = s[i] * fac;
    }

    if (MODE < 2) {
        // ---- agreement a[b,o] = sum_c u_hat[b,o,c] * v[b,c]; b_new = b_old + a
        float aout[8];
        #pragma unroll
        for (int o = 0; o < ONUM; ++o) {
            #pragma unroll
            for (int i = 0; i < 8; ++i) {
                const float t = halfsum16(U[o][i] * vj[i]);
                if (m16 == o) aout[i] = t;   // lane o of each half keeps it
            }
        }
        #pragma unroll
        for (int i = 0; i < 8; ++i) {
            const int b = bb0 + i + 8 * h;
            bmat[((size_t)b * RNUM + r) * ONUM + m16] = braw[i] + aout[i];
        }
    } else {
        // ---- final output v_j -> (B, R, C)
        #pragma unroll
        for (int i = 0; i < 8; ++i) {
            const int b = bb0 + i + 8 * h;
            out[((size_t)b * RNUM + r) * CNUM + m16] = vj[i];
        }
    }
}

// Per-(b,o): rowmax = max_r bmat[b,r,o]; denom = sum_r exp(bmat - rowmax).
// One block per b; thread t -> (o = t&15, r-stripe = t>>4).
__global__ __launch_bounds__(256) void softmax_reduce(
    const float* __restrict__ bmat,   // (B, R, O)
    float* __restrict__ rowmax,       // (B, O)
    float* __restrict__ denom)        // (B, O)
{
    __shared__ float sm[256];
    const int b  = blockIdx.x;
    const int t  = threadIdx.x;
    const int o  = t & 15;
    const int rs = t >> 4;            // 16 stripes over R

    float mx = -INFINITY;
    for (int r = rs; r < RNUM; r += 16)
        mx = fmaxf(mx, bmat[((size_t)b * RNUM + r) * ONUM + o]);
    sm[t] = mx;
    __syncthreads();
    for (int off = 128; off >= 16; off >>= 1) {
        if (t < off) sm[t] = fmaxf(sm[t], sm[t + off]);
        __syncthreads();
    }
    const float rm = sm[o];
    if (t < 16) rowmax[b * ONUM + t] = sm[t];
    __syncthreads();

    float se = 0.0f;
    for (int r = rs; r < RNUM; r += 16)
        se += __expf(bmat[((size_t)b * RNUM + r) * ONUM + o] - rm);
    sm[t] = se;
    __syncthreads();
    for (int off = 128; off >= 16; off >>= 1) {
        if (t < off) sm[t] += sm[t + off];
        __syncthreads();
    }
    if (t < 16) denom[b * ONUM + t] = sm[t];
}

extern "C" void kernel_launch(void* const* d_in, const int* in_sizes, int n_in,
                              void* d_out, int out_size, void* d_ws, size_t ws_size,
                              hipStream_t stream) {
    const float* x = (const float*)d_in[0];        // (128,1,1,16)
    const float* W = (const float*)d_in[1];        // (1,2048,16,16,16)
    float* out = (float*)d_out;                    // (128,2048,16)

    // Scratch layout in d_ws: bmat (B*R*O f32 = 16 MB) + rowmax + denom (8 KB each)
    float* bmat = (float*)d_ws;
    float* rowmax = bmat + (size_t)BATCH * RNUM * ONUM;
    float* denom  = rowmax + BATCH * ONUM;

    dim3 grid(RNUM), block(256);

    // iter 0: uniform softmax, produce b1
    caps_kernel<0><<<grid, block, 0, stream>>>(x, W, bmat, rowmax, denom, out);
    softmax_reduce<<<dim3(BATCH), block, 0, stream>>>(bmat, rowmax, denom);
    // iter 1: softmax(b1), produce b2 (in place)
    caps_kernel<1><<<grid, block, 0, stream>>>(x, W, bmat, rowmax, denom, out);
    softmax_reduce<<<dim3(BATCH), block, 0, stream>>>(bmat, rowmax, denom);
    // iter 2: softmax(b2), write v_j
    caps_kernel<2><<<grid, block, 0, stream>>>(x, W, bmat, rowmax, denom, out);
}